// SCSSystem_146028888299
// MI455X (gfx1250) — compile-verified
//
#include <hip/hip_runtime.h>
#include <hip/hip_bf16.h>

typedef float v2f __attribute__((ext_vector_type(2)));
typedef float v8f __attribute__((ext_vector_type(8)));

#define PSZ     8      // source/target patch size
#define HGT     256
#define WID     256
#define HP      32     // patches per dimension
#define PNUM    1024   // total patches
#define SDIM    64     // source patch elems (K)
#define TDIM    64     // target patch elems (N)
#define BATCH   128    // M
#define LSTRIDE 68     // padded LDS row stride (272B: 16B-aligned, bank-conflict-free)

// One block = one patch of one connection. 256 threads = 8 wave32s.
// Wave w computes batches [16w, 16w+16) x all 64 t-outputs via
// v_wmma_f32_16x16x4_f32, K accumulated in steps of 4 over s=0..63.
__global__ __launch_bounds__(256) void axonal_wmma_kernel(
    const float* __restrict__ spk_pfc,
    const float* __restrict__ spk_acc,
    const float* __restrict__ gates_pfc_acc,
    const float* __restrict__ T_pfc_acc,
    const float* __restrict__ gates_acc_pfc,
    const float* __restrict__ T_acc_pfc,
    float* __restrict__ out)
{
    __shared__ float As[BATCH * LSTRIDE];  // unfolded spikes: [b][s]
    __shared__ float Ts[TDIM * LSTRIDE];   // transform tile:  [t][s]

    const int p    = blockIdx.x;
    const int conn = blockIdx.y;  // 0 -> axon_pfc (spk_acc,T_acc_pfc), 1 -> axon_acc (spk_pfc,T_pfc_acc)

    const float* spk   = (conn == 0) ? spk_acc       : spk_pfc;
    const float* T     = (conn == 0) ? T_acc_pfc     : T_pfc_acc;
    const float* gates = (conn == 0) ? gates_acc_pfc : gates_pfc_acc;
    float* outc = out + (size_t)conn * (size_t)BATCH * HGT * WID;

    const int py  = p >> 5;   // patch row
    const int px  = p & 31;   // patch col
    const int tid = threadIdx.x;

    // ---- Stage T[p] (64x64 f32 = 16 KB) into LDS, row-major [t][s] with pad.
    {
        const float4* T4 = reinterpret_cast<const float4*>(T + (size_t)p * (TDIM * SDIM));
        #pragma unroll
        for (int i = 0; i < 4; ++i) {
            int idx = tid + i * 256;          // 1024 float4s total
            int t   = idx >> 4;               // 16 float4 per row of 64
            int s   = (idx & 15) << 2;
            float4 v = T4[idx];
            *reinterpret_cast<float4*>(&Ts[t * LSTRIDE + s]) = v;
        }
    }

    // ---- Stage unfolded spikes A (128x64 f32 = 32 KB) into LDS [b][s].
    // s = sy*8+sx maps to global (py*8+sy, px*8+sx); each (b,sy) row = 2 float4.
    {
        const float4* S4 = reinterpret_cast<const float4*>(spk);
        #pragma unroll
        for (int i = 0; i < 8; ++i) {
            int idx  = tid + i * 256;         // 2048 float4s total
            int b    = idx >> 4;
            int r    = idx & 15;
            int sy   = r >> 1;
            int half = r & 1;
            int gf   = (b * HGT + py * PSZ + sy) * WID + px * PSZ + half * 4;
            float4 v = S4[gf >> 2];
            *reinterpret_cast<float4*>(&As[b * LSTRIDE + sy * PSZ + half * 4]) = v;
        }
    }
    __syncthreads();

    const int wave = tid >> 5;   // 0..7 (wave32)
    const int lane = tid & 31;
    const int hi   = lane >> 4;  // K-half selector for A/B frags, M-half for C/D
    const int nl   = lane & 15;  // M index (A), N index (B, C/D)
    const int mb   = wave * 16;  // batch tile base

    v8f acc[4] = {v8f{}, v8f{}, v8f{}, v8f{}};

    const float* arow = &As[(mb + nl) * LSTRIDE + 2 * hi];
    const float* b0r  = &Ts[( 0 + nl) * LSTRIDE + 2 * hi];
    const float* b1r  = &Ts[(16 + nl) * LSTRIDE + 2 * hi];
    const float* b2r  = &Ts[(32 + nl) * LSTRIDE + 2 * hi];
    const float* b3r  = &Ts[(48 + nl) * LSTRIDE + 2 * hi];

    #pragma unroll
    for (int k0 = 0; k0 < SDIM; k0 += 4) {
        // A frag: lanes 0-15 -> K = {k0,k0+1}, lanes 16-31 -> K = {k0+2,k0+3}; M = lane&15
        v2f a  = *reinterpret_cast<const v2f*>(arow + k0);
        // B frag: B[k][n] = T[n][k]; same K-pair split by lane half; N = lane&15
        v2f b0 = *reinterpret_cast<const v2f*>(b0r + k0);
        v2f b1 = *reinterpret_cast<const v2f*>(b1r + k0);
        v2f b2 = *reinterpret_cast<const v2f*>(b2r + k0);
        v2f b3 = *reinterpret_cast<const v2f*>(b3r + k0);
        acc[0] = __builtin_amdgcn_wmma_f32_16x16x4_f32(false, a, false, b0, (short)0, acc[0], false, false);
        acc[1] = __builtin_amdgcn_wmma_f32_16x16x4_f32(false, a, false, b1, (short)0, acc[1], false, false);
        acc[2] = __builtin_amdgcn_wmma_f32_16x16x4_f32(false, a, false, b2, (short)0, acc[2], false, false);
        acc[3] = __builtin_amdgcn_wmma_f32_16x16x4_f32(false, a, false, b3, (short)0, acc[3], false, false);
    }

    // ---- Fused gate scale + fold store: out[b, py*8+ty, px*8+tx] = acc * g
    const float g = gates[p];
    #pragma unroll
    for (int nt = 0; nt < 4; ++nt) {
        int t  = nt * 16 + nl;       // C/D: N = lane&15 within this 16-wide tile
        int ty = t >> 3;
        int tx = t & 7;
        #pragma unroll
        for (int j = 0; j < 8; ++j) {
            int b = mb + j + 8 * hi; // C/D: VGPR j holds rows M=j (lo lanes) / M=j+8 (hi lanes)
            outc[(size_t)(b * HGT + py * PSZ + ty) * WID + px * PSZ + tx] = acc[nt][j] * g;
        }
    }
}

extern "C" void kernel_launch(void* const* d_in, const int* in_sizes, int n_in,
                              void* d_out, int out_size, void* d_ws, size_t ws_size,
                              hipStream_t stream) {
    const float* spk_pfc       = (const float*)d_in[0];
    const float* spk_acc       = (const float*)d_in[1];
    const float* gates_pfc_acc = (const float*)d_in[2];
    const float* T_pfc_acc     = (const float*)d_in[3];
    const float* gates_acc_pfc = (const float*)d_in[4];
    const float* T_acc_pfc     = (const float*)d_in[5];
    float* out = (float*)d_out;

    dim3 grid(PNUM, 2);   // 1024 patches x 2 connections
    dim3 block(256);      // 8 wave32s
    axonal_wmma_kernel<<<grid, block, 0, stream>>>(
        spk_pfc, spk_acc, gates_pfc_acc, T_pfc_acc, gates_acc_pfc, T_acc_pfc, out);
}